// GCN_35579509080730
// MI455X (gfx1250) — compile-verified
//
#include <hip/hip_runtime.h>
#include <hip/hip_bf16.h>

#define N_NODES 100000
#define N_EDGES 1200000
#define FEATS   64
#define N_TILES (N_NODES / 16)   // 6250, exact

typedef __attribute__((ext_vector_type(2))) float v2f;
typedef __attribute__((ext_vector_type(8))) float v8f;

// ---------------------------------------------------------------------------
// Kernel 0: zero the workspace accumulators (sum[N*64] followed by deg[N]).
// ---------------------------------------------------------------------------
__global__ void gcn_zero_ws(float* __restrict__ ws, int n) {
    int i = blockIdx.x * blockDim.x + threadIdx.x;
    if (i < n) ws[i] = 0.0f;
}

// ---------------------------------------------------------------------------
// Kernel 1: per-edge scatter. 64 threads handle one edge: coalesced 256B
// gather of the source row, then f32 atomic adds into the L2-resident
// accumulator. Lane 0 of each edge-group bumps the degree counter.
// ---------------------------------------------------------------------------
__global__ __launch_bounds__(256) void gcn_scatter(
    const float* __restrict__ feature,
    const int*   __restrict__ edge_src,
    const int*   __restrict__ edge_dst,
    float* __restrict__ sum,
    float* __restrict__ deg)
{
    int t = blockIdx.x * 256 + threadIdx.x;
    int e = t >> 6;          // edge index
    int c = t & 63;          // feature index
    if (e >= N_EDGES) return;
    int s = edge_src[e];
    int d = edge_dst[e];
    float v = feature[(size_t)s * FEATS + c];
    atomicAdd(&sum[(size_t)d * FEATS + c], v);
    if (c == 0) atomicAdd(&deg[d], 1.0f);
}

// ---------------------------------------------------------------------------
// Kernel 2: fused mean + (h @ W^T + b) + ReLU using V_WMMA_F32_16X16X4_F32.
// One wave computes a 16-node x 64-output tile (four 16x16 accumulators).
// W staged to padded LDS (stride 65) by the whole block; each wave stages its
// own mean-scaled 16x64 A-tile into wave-private padded LDS.
// ---------------------------------------------------------------------------
__global__ __launch_bounds__(256) void gcn_mean_linear_relu(
    const float* __restrict__ sum,
    const float* __restrict__ deg,
    const float* __restrict__ W,     // [64 out][64 in]
    const float* __restrict__ bias,  // [64]
    float* __restrict__ out)         // [N][64]
{
    __shared__ float sW[64 * 65];          // padded: row stride 65
    __shared__ float sA[8][16 * 65];       // per-wave padded tile

    // Cooperative load of W into LDS (whole block).
    for (int i = threadIdx.x; i < 64 * 64; i += 256) {
        int r = i >> 6;
        int c = i & 63;
        sW[r * 65 + c] = W[i];
    }
    __syncthreads();

    const int wave = threadIdx.x >> 5;
    const int lane = threadIdx.x & 31;
    const int tile = blockIdx.x * 8 + wave;
    if (tile >= N_TILES) return;           // uniform per-wave guard
    const int nodeBase = tile * 16;

    // Stage this wave's 16x64 tile, scaled by 1/max(deg,1), into LDS.
    float* myA = sA[wave];
    const float* gsum = sum + (size_t)nodeBase * FEATS;
    #pragma unroll
    for (int it = 0; it < 32; ++it) {
        int idx = it * 32 + lane;          // 0..1023, coalesced
        int r = idx >> 6;
        int c = idx & 63;
        float dg  = deg[nodeBase + r];
        float inv = 1.0f / fmaxf(dg, 1.0f);
        myA[r * 65 + c] = gsum[idx] * inv;
    }
    // Wave-private LDS slab: no barrier needed, compiler waits on DScnt.

    const int row  = lane & 15;            // A row / B col for this lane
    const int kb   = (lane >> 4) * 2;      // K sub-offset: 0 or 2
    const int half = lane >> 4;

    v8f acc0 = {}, acc1 = {}, acc2 = {}, acc3 = {};

    #pragma unroll
    for (int k0 = 0; k0 < 64; k0 += 4) {
        v2f a;
        a.x = myA[row * 65 + k0 + kb];
        a.y = myA[row * 65 + k0 + kb + 1];

        v2f b0, b1, b2, b3;
        b0.x = sW[(row      ) * 65 + k0 + kb];
        b0.y = sW[(row      ) * 65 + k0 + kb + 1];
        b1.x = sW[(16 + row ) * 65 + k0 + kb];
        b1.y = sW[(16 + row ) * 65 + k0 + kb + 1];
        b2.x = sW[(32 + row ) * 65 + k0 + kb];
        b2.y = sW[(32 + row ) * 65 + k0 + kb + 1];
        b3.x = sW[(48 + row ) * 65 + k0 + kb];
        b3.y = sW[(48 + row ) * 65 + k0 + kb + 1];

        acc0 = __builtin_amdgcn_wmma_f32_16x16x4_f32(
            false, a, false, b0, (short)0, acc0, false, false);
        acc1 = __builtin_amdgcn_wmma_f32_16x16x4_f32(
            false, a, false, b1, (short)0, acc1, false, false);
        acc2 = __builtin_amdgcn_wmma_f32_16x16x4_f32(
            false, a, false, b2, (short)0, acc2, false, false);
        acc3 = __builtin_amdgcn_wmma_f32_16x16x4_f32(
            false, a, false, b3, (short)0, acc3, false, false);
    }

    // Epilogue: bias + ReLU + store. C/D layout: VGPR v holds rows v (lanes
    // 0-15) and v+8 (lanes 16-31); column = lane&15.
    const int col   = lane & 15;
    const int mbase = 8 * half;
    float bb0 = bias[ 0 + col];
    float bb1 = bias[16 + col];
    float bb2 = bias[32 + col];
    float bb3 = bias[48 + col];
    #pragma unroll
    for (int v = 0; v < 8; ++v) {
        size_t rowOff = (size_t)(nodeBase + mbase + v) * FEATS;
        out[rowOff +  0 + col] = fmaxf(acc0[v] + bb0, 0.0f);
        out[rowOff + 16 + col] = fmaxf(acc1[v] + bb1, 0.0f);
        out[rowOff + 32 + col] = fmaxf(acc2[v] + bb2, 0.0f);
        out[rowOff + 48 + col] = fmaxf(acc3[v] + bb3, 0.0f);
    }
}

// ---------------------------------------------------------------------------
extern "C" void kernel_launch(void* const* d_in, const int* in_sizes, int n_in,
                              void* d_out, int out_size, void* d_ws, size_t ws_size,
                              hipStream_t stream) {
    const float* feature  = (const float*)d_in[0];
    const int*   edge_src = (const int*)  d_in[1];
    const int*   edge_dst = (const int*)  d_in[2];
    const float* W        = (const float*)d_in[3];
    const float* bias     = (const float*)d_in[4];
    float*       out      = (float*)d_out;

    float* sum = (float*)d_ws;                       // N_NODES * 64 floats
    float* deg = sum + (size_t)N_NODES * FEATS;      // N_NODES floats

    // 1) Zero accumulators.
    int zn = N_NODES * FEATS + N_NODES;
    gcn_zero_ws<<<(zn + 255) / 256, 256, 0, stream>>>((float*)d_ws, zn);

    // 2) Edge scatter: 4 edges per 256-thread block.
    int scatterBlocks = (N_EDGES * 64 + 255) / 256;  // 300000
    gcn_scatter<<<scatterBlocks, 256, 0, stream>>>(feature, edge_src, edge_dst,
                                                   sum, deg);

    // 3) Fused mean + linear + ReLU: 8 node-tiles (waves) per block.
    int gemmBlocks = (N_TILES + 7) / 8;              // 782
    gcn_mean_linear_relu<<<gemmBlocks, 256, 0, stream>>>(sum, deg, W, bias, out);
}